// FOCAL_58368605553242
// MI455X (gfx1250) — compile-verified
//
#include <hip/hip_runtime.h>

typedef float v2f __attribute__((ext_vector_type(2)));
typedef float v8f __attribute__((ext_vector_type(8)));
typedef unsigned int u32x4 __attribute__((ext_vector_type(4)));
typedef int i32x4 __attribute__((ext_vector_type(4)));
typedef int i32x8 __attribute__((ext_vector_type(8)));

#define MB   16      // tags per batch
#define NB   120     // patches per batch
#define DB   256     // embedding dim
#define SD   260     // LDS row stride for z/h/e (260 % 64 == 4 -> conflict-free frag loads)
#define TS   132     // LDS row stride for log_K / T (132 % 64 == 4)
#define NITER 30

// eps=0.1, tau=1.0 -> -1/eps = -10, scale = tau/(tau+eps) = 1/1.1
#define INV_EPS   10.0f
#define UOT_SCALE 0.90909090909090909f
#define LOG_A    (-2.7725887f)   // -ln(16)
#define LOG_B    (-4.7874917f)   // -ln(120)

// ---- Tensor Data Mover: DMA a (rows x 256 f32) row-major tile into LDS with
// ---- +4 DWORD row padding (LDS row stride becomes 260 floats). ISA 08 §8.
__device__ __forceinline__ void tdm_load_tile_pad260(const float* gsrc,
                                                     unsigned lds_byte_off,
                                                     unsigned rows) {
  unsigned long long ga = (unsigned long long)(uintptr_t)gsrc;

  u32x4 g0;
  g0[0] = 1u;                                   // count=1 (valid), user mode, no gather
  g0[1] = lds_byte_off;                         // lds_addr [63:32]
  g0[2] = (unsigned)(ga & 0xFFFFFFFFu);         // global_addr [95:64]
  g0[3] = (unsigned)((ga >> 32) & 0x01FFFFFFu)  // global_addr [120:96]
          | 0x80000000u;                        // type=2 ("image") [127:126]

  i32x8 g1;
  g1[0] = (int)((2u << 16)                      // data_size = 4 bytes
              | (1u << 20)                      // pad_enable
              | (7u << 22)                      // pad_interval: 256 DWORDs
              | (3u << 25));                    // pad_amount:   4 DWORDs
  g1[1] = (int)(DB << 16);                      // [47:32] atomic_barrier=0 | tensor_dim0[15:0]=256
  g1[2] = (int)(rows << 16);                    // tensor_dim0[31:16]=0 | tensor_dim1[15:0]=rows
  g1[3] = (int)(DB << 16);                      // tensor_dim1[31:16]=0 | tile_dim0=256
  g1[4] = (int)rows;                            // tile_dim1=rows | tile_dim2=0
  g1[5] = DB;                                   // tensor_dim0_stride[31:0]=256
  g1[6] = 0;                                    // stride0[47:32]=0 | stride1[15:0]=0
  g1[7] = 0;                                    // tensor_dim1_stride[47:16]=0

  i32x4 gz4 = {0, 0, 0, 0};                     // groups 2/3 unused (2D tile)
  i32x8 gz8 = {0, 0, 0, 0, 0, 0, 0, 0};
  __builtin_amdgcn_tensor_load_to_lds(g0, g1, gz4, gz4, gz8, 0);
}

__global__ __launch_bounds__(256) void uot_sinkhorn_kernel(
    const float* __restrict__ tag,
    const float* __restrict__ patch,
    float* __restrict__ out) {
  __shared__ __align__(16) float zs[MB * SD];    // normalized tags, later reused for e
  __shared__ __align__(16) float hs[NB * SD];    // normalized patches
  __shared__ __align__(16) float Ts[MB * TS];    // log_K, later T_hat
  __shared__ float log_u[MB];
  __shared__ float log_v[NB];
  __shared__ float rnorm[MB + NB];

  const int tid  = threadIdx.x;
  const int b    = blockIdx.x;
  const int lane = tid & 31;
  const int wave = tid >> 5;       // 0..7
  const int l16  = lane & 15;      // row/col within frag
  const int hi   = lane >> 4;      // 0/1: K offset 0 or 2

  // ---------------- Phase 0: TDM-stage inputs into LDS (padded rows) ----------------
  if (wave == 0) {
    tdm_load_tile_pad260(patch + (size_t)b * (NB * DB),
                         (unsigned)(uintptr_t)hs, NB);
    __builtin_amdgcn_s_wait_tensorcnt(0);
  } else if (wave == 1) {
    tdm_load_tile_pad260(tag + (size_t)b * (MB * DB),
                         (unsigned)(uintptr_t)zs, MB);
    __builtin_amdgcn_s_wait_tensorcnt(0);
  }
  if (tid < NB) log_v[tid] = 0.0f;   // log_v0 = zeros
  __syncthreads();

  // ---------------- l2-normalize all 136 rows (16 lanes per row) ----------------
  {
    const int sub = tid & 15;
    for (int r = tid >> 4; r < MB + NB; r += 16) {
      const float* row = (r < MB) ? (zs + r * SD) : (hs + (r - MB) * SD);
      float s = 0.0f;
      for (int j = sub; j < DB; j += 16) { float x = row[j]; s += x * x; }
      s += __shfl_xor(s, 1, 16);
      s += __shfl_xor(s, 2, 16);
      s += __shfl_xor(s, 4, 16);
      s += __shfl_xor(s, 8, 16);
      if (sub == 0) rnorm[r] = 1.0f / fmaxf(sqrtf(s), 1e-12f);
    }
  }
  __syncthreads();
  for (int idx = tid; idx < (MB + NB) * DB; idx += 256) {
    int r = idx >> 8, c = idx & 255;
    float* row = (r < MB) ? (zs + r * SD) : (hs + (r - MB) * SD);
    row[c] *= rnorm[r];
  }
  __syncthreads();

  // ---------------- GEMM1: log_K[m][n] = (z.h - 1) / eps, via f32 WMMA ----------------
  {
    const int nc = wave * 16 + l16;                 // this lane's output column
    const int nr = (nc < NB) ? nc : (NB - 1);       // clamp B row for pad lanes
    const float* za = zs + l16 * SD + 2 * hi;       // A frag base: row m = l16
    const float* hb = hs + nr  * SD + 2 * hi;       // B frag base: row n = nc
    v8f acc = {};
    #pragma unroll 8
    for (int k = 0; k < DB; k += 4) {
      v2f a  = *(const v2f*)(za + k);
      v2f bb = *(const v2f*)(hb + k);
      acc = __builtin_amdgcn_wmma_f32_16x16x4_f32(
          false, a, false, bb, (short)0, acc, false, false);
    }
    if (nc < NB) {
      #pragma unroll
      for (int r = 0; r < 8; ++r)
        Ts[(r + 8 * hi) * TS + nc] = (acc[r] - 1.0f) * INV_EPS;
    }
  }
  __syncthreads();

  // ---------------- Sinkhorn: 30 log-domain iterations ----------------
  for (int it = 0; it < NITER; ++it) {
    // u-update: log_u[m] = log_a - logsumexp_n(log_K[m][n] + log_v[n])
    {
      const int m = tid >> 4, sub = tid & 15;
      const float* row = Ts + m * TS;
      float mx = -3.4e38f;
      for (int n = sub; n < NB; n += 16) mx = fmaxf(mx, row[n] + log_v[n]);
      mx = fmaxf(mx, __shfl_xor(mx, 1, 16));
      mx = fmaxf(mx, __shfl_xor(mx, 2, 16));
      mx = fmaxf(mx, __shfl_xor(mx, 4, 16));
      mx = fmaxf(mx, __shfl_xor(mx, 8, 16));
      float s = 0.0f;
      for (int n = sub; n < NB; n += 16) s += __expf(row[n] + log_v[n] - mx);
      s += __shfl_xor(s, 1, 16);
      s += __shfl_xor(s, 2, 16);
      s += __shfl_xor(s, 4, 16);
      s += __shfl_xor(s, 8, 16);
      if (sub == 0) log_u[m] = LOG_A - (mx + __logf(s));
    }
    __syncthreads();
    // v-update: log_v[n] = scale * (log_b - logsumexp_m(log_K[m][n] + log_u[m]))
    if (tid < NB) {
      const int n = tid;
      float vals[MB];
      float mx = -3.4e38f;
      #pragma unroll
      for (int m = 0; m < MB; ++m) {
        vals[m] = Ts[m * TS + n] + log_u[m];
        mx = fmaxf(mx, vals[m]);
      }
      float s = 0.0f;
      #pragma unroll
      for (int m = 0; m < MB; ++m) s += __expf(vals[m] - mx);
      log_v[n] = UOT_SCALE * (LOG_B - (mx + __logf(s)));
    }
    __syncthreads();
  }

  // ---------------- T_hat = exp(log_u + log_K + log_v) * m (in place) ----------------
  for (int idx = tid; idx < MB * NB; idx += 256) {
    int m = idx / NB, n = idx - m * NB;
    Ts[m * TS + n] = __expf(log_u[m] + Ts[m * TS + n] + log_v[n]) * 16.0f;
  }
  __syncthreads();

  // ---------------- GEMM2: e[m][d] = sum_n T[m][n] * h[n][d], via f32 WMMA ----------------
  // K = 120 exactly (multiple of 4): no padding needed.
  for (int tt = 0; tt < 2; ++tt) {
    const int dc = (wave + 8 * tt) * 16 + l16;      // output column d
    const float* ta = Ts + l16 * TS + 2 * hi;       // A frag base: row m = l16
    v8f acc = {};
    #pragma unroll 6
    for (int k = 0; k < NB; k += 4) {
      int k0 = k + 2 * hi;
      v2f a  = *(const v2f*)(ta + k);
      v2f bb;
      bb.x = hs[k0 * SD + dc];
      bb.y = hs[(k0 + 1) * SD + dc];
      acc = __builtin_amdgcn_wmma_f32_16x16x4_f32(
          false, a, false, bb, (short)0, acc, false, false);
    }
    #pragma unroll
    for (int r = 0; r < 8; ++r)
      zs[(r + 8 * hi) * SD + dc] = acc[r];          // reuse zs as e buffer
  }
  __syncthreads();

  // ---------------- final l2-normalize of e and store ----------------
  {
    const int m = tid >> 4, sub = tid & 15;
    float s = 0.0f;
    for (int j = sub; j < DB; j += 16) { float x = zs[m * SD + j]; s += x * x; }
    s += __shfl_xor(s, 1, 16);
    s += __shfl_xor(s, 2, 16);
    s += __shfl_xor(s, 4, 16);
    s += __shfl_xor(s, 8, 16);
    if (sub == 0) rnorm[m] = 1.0f / fmaxf(sqrtf(s), 1e-12f);
  }
  __syncthreads();
  float* ob = out + (size_t)b * (MB * DB);
  for (int idx = tid; idx < MB * DB; idx += 256) {
    int m = idx >> 8;
    ob[idx] = zs[m * SD + (idx & 255)] * rnorm[m];
  }
}

extern "C" void kernel_launch(void* const* d_in, const int* in_sizes, int n_in,
                              void* d_out, int out_size, void* d_ws, size_t ws_size,
                              hipStream_t stream) {
  const float* tag   = (const float*)d_in[0];   // (B, 16, 256) f32
  const float* patch = (const float*)d_in[1];   // (B, 120, 256) f32
  float* out = (float*)d_out;                   // (B, 16, 256) f32
  const int B = in_sizes[0] / (MB * DB);
  uot_sinkhorn_kernel<<<B, 256, 0, stream>>>(tag, patch, out);
}